// GRVQ_2559800508671
// MI455X (gfx1250) — compile-verified
//
#include <hip/hip_runtime.h>
#include <hip/hip_bf16.h>
#include <cstddef>

// ---------------- problem constants ----------------
constexpr int Bv  = 8;
constexpr int DIMv = 512;
constexpr int Lv  = 4096;
constexpr int Gv  = 2;
constexpr int NQv = 8;
constexpr int CSv = 1024;
constexpr int DGv = 256;          // DIM / G
constexpr int TOKv = Bv * Lv;     // 32768 tokens

// output layout (floats) in d_out
constexpr size_t Q_OFF   = 0;                                  // q: [B, DIM, L]
constexpr size_t IDX_OFF = (size_t)Bv * DIMv * Lv;             // indices: [G, B, L, NQ]
constexpr size_t LOSS_OFF = IDX_OFF + (size_t)Gv * TOKv * NQv; // losses: [G, NQ]

typedef __attribute__((ext_vector_type(2))) float v2f;
typedef __attribute__((ext_vector_type(4))) float v4f;
typedef __attribute__((ext_vector_type(8))) float v8f;
typedef __attribute__((ext_vector_type(4))) int   v4i;

// gfx1250 async global->LDS path (guarded so either toolchain compiles)
#if defined(__has_builtin)
#if __has_builtin(__builtin_amdgcn_global_load_async_to_lds_b128)
#define USE_ASYNC_LDS 1
#endif
#endif

__device__ __forceinline__ void wait_async_lds0() {
#if defined(USE_ASYNC_LDS)
#if __has_builtin(__builtin_amdgcn_s_wait_asynccnt)
  __builtin_amdgcn_s_wait_asynccnt(0);
#else
  asm volatile("s_wait_asynccnt 0" ::: "memory");
#endif
#endif
}

__device__ __forceinline__ v2f mkv2(float a, float b) { v2f v; v.x = a; v.y = b; return v; }
__device__ __forceinline__ v2f vlo(v4f v) { return mkv2(v.x, v.y); }
__device__ __forceinline__ v2f vhi(v4f v) { return mkv2(v.z, v.w); }

// D(16x16,f32) = A(16x4,f32) * B(4x16,f32) + C  — CDNA5 V_WMMA_F32_16X16X4_F32
__device__ __forceinline__ v8f wmma4(v2f a, v2f b, v8f c) {
  return __builtin_amdgcn_wmma_f32_16x16x4_f32(false, a, false, b, (short)0, c, false, false);
}

// ---------------- c2 precompute + loss accumulator init ----------------
__global__ __launch_bounds__(256) void grvq_c2_kernel(const float* __restrict__ cb,
                                                      float* __restrict__ c2buf,
                                                      float* __restrict__ lossacc) {
  int r = blockIdx.x * 256 + threadIdx.x;           // NQ*G*CS = 16384 rows
  if (r < NQv * Gv * CSv) {
    const float* p = cb + (size_t)r * DGv;
    float s = 0.f;
    #pragma unroll 8
    for (int k = 0; k < DGv; ++k) s += p[k] * p[k];
    c2buf[r] = s;
  }
  if (blockIdx.x == 0 && threadIdx.x < Gv * NQv) lossacc[threadIdx.x] = 0.f;
}

__global__ __launch_bounds__(256) void grvq_zero_kernel(float* __restrict__ p, int n) {
  for (int i = blockIdx.x * blockDim.x + threadIdx.x; i < n; i += gridDim.x * blockDim.x)
    p[i] = 0.f;
}

// ---------------- conv1d (k=3, same) as 3 accumulated WMMA GEMMs ----------------
// h1[b][o][l] = sum_{i,tap} w[o][i][tap] * x[b][i][l+tap-1] + bias[o]
// Boundary handled via clamped address + 0/1 mask multiply (no exec divergence).
__global__ __launch_bounds__(256) void grvq_conv_kernel(const float* __restrict__ x,
                                                        const float* __restrict__ w,
                                                        const float* __restrict__ bias,
                                                        float* __restrict__ h1) {
  const int tid = threadIdx.x, wave = tid >> 5, lane = tid & 31;
  const int half = lane >> 4, l16 = lane & 15;
  const int t = blockIdx.x * 8 + wave;      // 8 tiles / block, 65536 tiles total
  const int b   = t >> 13;                  // 8192 tiles per batch (32 M-tiles * 256 N-tiles)
  const int rem = t & 8191;
  const int m0 = (rem >> 8) * 16;           // out-channel tile
  const int n0 = (rem & 255) * 16;          // position tile
  const int mrow = m0 + l16;

  v8f acc = {0.f,0.f,0.f,0.f,0.f,0.f,0.f,0.f};
  for (int tap = 0; tap < 3; ++tap) {
    const int lraw = n0 + l16 + tap - 1;
    const int lc = lraw < 0 ? 0 : (lraw >= Lv ? Lv - 1 : lraw);
    const float vf = (lraw == lc) ? 1.f : 0.f;     // 0 for padded positions
    const float* xcol = x + (size_t)b * DIMv * Lv + lc;
    const float* wrow = w + (size_t)mrow * DIMv * 3 + tap;
    #pragma unroll 8
    for (int kk = 0; kk < DIMv / 4; ++kk) {
      const int k = kk * 4 + half * 2;
      v2f a = mkv2(wrow[(size_t)k * 3], wrow[(size_t)(k + 1) * 3]);
      v2f bb = mkv2(xcol[(size_t)k * Lv] * vf, xcol[(size_t)(k + 1) * Lv] * vf);
      acc = wmma4(a, bb, acc);
    }
  }
  #pragma unroll
  for (int r = 0; r < 8; ++r) {
    const int m = m0 + r + half * 8;
    h1[((size_t)b * DIMv + m) * Lv + n0 + l16] = acc[r] + bias[m];
  }
}

// ---------------- pre-linear: resid[token][n] = sum_k h1[b][k][l] * pre_w[n][k] + pre_b[n]
// K reordered per lane-half: half h covers k in [256h, 256h+256) -> v4f B loads.
__global__ __launch_bounds__(256) void grvq_pre_kernel(const float* __restrict__ h1,
                                                       const float* __restrict__ pw,
                                                       const float* __restrict__ pb,
                                                       float* __restrict__ resid) {
  const int tid = threadIdx.x, wave = tid >> 5, lane = tid & 31;
  const int half = lane >> 4, l16 = lane & 15;
  const int t = blockIdx.x * 8 + wave;      // 65536 tiles: 2048 M-tiles x 32 N-tiles
  const int mt = t >> 5, n0 = (t & 31) * 16;

  const int tokA = mt * 16 + l16;
  const int bA = tokA >> 12, lA = tokA & (Lv - 1);
  const int kbase = half * (DIMv / 2);
  const float* ap = h1 + (size_t)bA * DIMv * Lv + lA;      // element k at ap[k*L]
  const float* bp = pw + (size_t)(n0 + l16) * DIMv + kbase;

  v8f acc = {0.f,0.f,0.f,0.f,0.f,0.f,0.f,0.f};
  #pragma unroll 8
  for (int j = 0; j < DIMv / 8; ++j) {      // 64 iters, 2 WMMA each
    const int k = kbase + j * 4;
    v4f b4 = *(const v4f*)(bp + j * 4);
    v2f a0 = mkv2(ap[(size_t)k * Lv], ap[(size_t)(k + 1) * Lv]);
    v2f a1 = mkv2(ap[(size_t)(k + 2) * Lv], ap[(size_t)(k + 3) * Lv]);
    acc = wmma4(a0, vlo(b4), acc);
    acc = wmma4(a1, vhi(b4), acc);
  }
  const int col = n0 + l16;
  const float bias = pb[col];
  #pragma unroll
  for (int r = 0; r < 8; ++r) {
    const int token = mt * 16 + r + half * 8;
    resid[(size_t)token * DIMv + col] = acc[r] + bias;
  }
}

// ---------------- one residual-VQ stage (both groups) ----------------
// Codebook tiles (16 codes x 256 dims) staged into LDS once per workgroup
// (async global->LDS on gfx1250, double-buffered). K reordered per lane-half
// (half h covers k in [128h, 128h+128) of the group) so B-fragments for two
// consecutive WMMAs are one 16B ds_load_b128 and A-fragments are 32 contiguous
// global b128 loads held in registers for the whole stage.
constexpr int LROW = 260;                         // padded row: 1040B, 16B aligned
__global__ __launch_bounds__(256) void grvq_stage_kernel(float* __restrict__ resid,
                                                         float* __restrict__ qout,
                                                         const float* __restrict__ cb,
                                                         const float* __restrict__ c2buf,
                                                         float* __restrict__ lossacc,
                                                         float* __restrict__ idx_out,
                                                         int q) {
  __shared__ float cbtile[2][16 * LROW];          // 2 x 16.25KB double buffer

  const int tid = threadIdx.x, wave = tid >> 5, lane = tid & 31;
  const int half = lane >> 4, l16 = lane & 15;
  const int tok0 = blockIdx.x * 128 + wave * 16;  // 16 tokens per wave
  const int srow = tid >> 4;                      // staging: row 0..15
  const int sseg = tid & 15;                      // staging: 16B segment base

  for (int g = 0; g < Gv; ++g) {
    const float* cbg = cb + (size_t)(q * Gv + g) * CSv * DGv;
    const float* c2g = c2buf + (size_t)(q * Gv + g) * CSv;

    // stage one 16-code tile into LDS buffer `buf` (straight row copy)
    auto stage_tile = [&](int ct, int buf) {
      const float* gp = cbg + ((size_t)(ct * 16 + srow)) * DGv;
      float* lp = &cbtile[buf][srow * LROW];
      #pragma unroll
      for (int j = 0; j < 4; ++j) {
        const int seg = sseg + j * 16;            // 64 x 16B per row
#if defined(USE_ASYNC_LDS)
        __builtin_amdgcn_global_load_async_to_lds_b128(
            (v4i*)(gp + seg * 4), (v4i*)(lp + seg * 4), 0, 0);
#else
        *(v4f*)(lp + seg * 4) = *(const v4f*)(gp + seg * 4);
#endif
      }
    };

    // Residual A-fragments (K-reordered): 32 x v4f = 128 VGPRs.
    v4f areg[32];
    {
      const float* ar = resid + (size_t)(tok0 + l16) * DIMv + g * DGv + half * 128;
      #pragma unroll
      for (int j = 0; j < 32; ++j) areg[j] = *(const v4f*)(ar + j * 4);
    }

    float mval[8]; int midx[8];
    #pragma unroll
    for (int r = 0; r < 8; ++r) { mval[r] = 3.4e38f; midx[r] = 0; }

    stage_tile(0, 0);
    wait_async_lds0();
    __syncthreads();

    for (int ct = 0; ct < CSv / 16; ++ct) {       // 64 code tiles
      if (ct + 1 < CSv / 16) stage_tile(ct + 1, (ct + 1) & 1);

      const float* bl = &cbtile[ct & 1][l16 * LROW + half * 128];
      v8f acc = {0.f,0.f,0.f,0.f,0.f,0.f,0.f,0.f};
      #pragma unroll
      for (int j = 0; j < 32; ++j) {
        v4f b4 = *(const v4f*)(bl + j * 4);       // ds_load_b128, optimal banking
        acc = wmma4(vlo(areg[j]), vlo(b4), acc);
        acc = wmma4(vhi(areg[j]), vhi(b4), acc);
      }
      const int n = ct * 16 + l16;
      const float c2n = c2g[n];
      #pragma unroll
      for (int r = 0; r < 8; ++r) {
        const float s = c2n - 2.0f * acc[r];      // ||r||^2 term constant over codes
        if (s < mval[r]) { mval[r] = s; midx[r] = n; }
      }
      wait_async_lds0();
      __syncthreads();
    }

    // argmin across the 16 lanes of each half (row m = r + 8*half)
    #pragma unroll
    for (int r = 0; r < 8; ++r) {
      for (int off = 8; off > 0; off >>= 1) {
        const float ov = __shfl_xor(mval[r], off, 32);
        const int   oi = __shfl_xor(midx[r], off, 32);
        if (ov < mval[r] || (ov == mval[r] && oi < midx[r])) { mval[r] = ov; midx[r] = oi; }
      }
    }

    // write indices: [G, B, L, NQ] (token = b*L + l)
    if (l16 == 0) {
      #pragma unroll
      for (int r = 0; r < 8; ++r) {
        const int token = tok0 + r + 8 * half;
        idx_out[((size_t)g * TOKv + token) * NQv + q] = (float)midx[r];
      }
    }

    // broadcast winners for all 16 rows
    int idxm[16];
    #pragma unroll
    for (int r = 0; r < 8; ++r) {
      idxm[r]     = __shfl(midx[r], 0, 32);
      idxm[r + 8] = __shfl(midx[r], 16, 32);
    }

    // gather code rows, update residual/qout, accumulate commit loss
    float lsum = 0.f;
    #pragma unroll
    for (int m = 0; m < 16; ++m) {
      const int token = tok0 + m;
      float* R  = resid + (size_t)token * DIMv + g * DGv;
      float* Qo = qout  + (size_t)token * DIMv + g * DGv;
      const float* Cr = cbg + (size_t)idxm[m] * DGv;
      for (int k = lane; k < DGv; k += 32) {
        const float qv = Cr[k];
        const float rv = R[k] - qv;               // new residual
        R[k]  = rv;
        Qo[k] += qv;                               // straight-through == quant
        lsum  += rv * rv;                          // (residual - quant)^2
      }
    }
    for (int off = 16; off > 0; off >>= 1) lsum += __shfl_xor(lsum, off, 32);
    if (lane == 0) atomicAdd(lossacc + g * NQv + q, lsum);
    __syncthreads();                               // group g+1 reuses LDS buffers
  }
}

// ---------------- post-linear + transpose to [B, DIM, L]
// Both operands contiguous: K reordered per lane-half -> v4f loads for A and B.
__global__ __launch_bounds__(256) void grvq_post_kernel(const float* __restrict__ qout,
                                                        const float* __restrict__ pw,
                                                        const float* __restrict__ pb,
                                                        float* __restrict__ out) {
  const int tid = threadIdx.x, wave = tid >> 5, lane = tid & 31;
  const int half = lane >> 4, l16 = lane & 15;
  const int t = blockIdx.x * 8 + wave;
  const int mt = t >> 5, n0 = (t & 31) * 16;

  const int tokA = mt * 16 + l16;
  const int kbase = half * (DIMv / 2);
  const float* ap = qout + (size_t)tokA * DIMv + kbase;
  const float* bp = pw + (size_t)(n0 + l16) * DIMv + kbase;

  v8f acc = {0.f,0.f,0.f,0.f,0.f,0.f,0.f,0.f};
  #pragma unroll 8
  for (int j = 0; j < DIMv / 8; ++j) {      // 64 iters, 2 WMMA each
    v4f a4 = *(const v4f*)(ap + j * 4);
    v4f b4 = *(const v4f*)(bp + j * 4);
    acc = wmma4(vlo(a4), vlo(b4), acc);
    acc = wmma4(vhi(a4), vhi(b4), acc);
  }
  const int col = n0 + l16;
  const float bias = pb[col];
  #pragma unroll
  for (int r = 0; r < 8; ++r) {
    const int token = mt * 16 + r + half * 8;
    const int bO = token >> 12, lO = token & (Lv - 1);
    out[((size_t)bO * DIMv + col) * Lv + lO] = acc[r] + bias;
  }
}

__global__ void grvq_loss_kernel(const float* __restrict__ lossacc, float* __restrict__ out) {
  if (threadIdx.x < Gv * NQv)
    out[threadIdx.x] = lossacc[threadIdx.x] * (1.0f / ((float)Bv * Lv * DGv));
}

extern "C" void kernel_launch(void* const* d_in, const int* in_sizes, int n_in,
                              void* d_out, int out_size, void* d_ws, size_t ws_size,
                              hipStream_t stream) {
  (void)in_sizes; (void)n_in; (void)out_size; (void)ws_size;
  const float* x    = (const float*)d_in[0];
  const float* cew  = (const float*)d_in[1];
  const float* ceb  = (const float*)d_in[2];
  const float* prew = (const float*)d_in[3];
  const float* preb = (const float*)d_in[4];
  const float* cb   = (const float*)d_in[5];
  const float* pow_ = (const float*)d_in[6];
  const float* pob  = (const float*)d_in[7];
  // d_in[8], d_in[9]: decoder conv — result discarded by reference, skipped.

  float* out = (float*)d_out;
  float* ws  = (float*)d_ws;

  constexpr size_t NBD = (size_t)Bv * DIMv * Lv;   // 16,777,216 floats
  float* h1    = ws;            // conv output [B, DIM, L]
  float* qout  = ws;            // aliases h1 (h1 dead after pre-linear)
  float* resid = ws + NBD;      // [TOK, DIM]
  float* c2    = ws + 2 * NBD;  // [NQ, G, CS]
  float* lossA = c2 + (size_t)NQv * Gv * CSv;      // [G, NQ]

  grvq_c2_kernel  <<<64,   256, 0, stream>>>(cb, c2, lossA);
  grvq_conv_kernel<<<8192, 256, 0, stream>>>(x, cew, ceb, h1);
  grvq_pre_kernel <<<8192, 256, 0, stream>>>(h1, prew, preb, resid);
  grvq_zero_kernel<<<4096, 256, 0, stream>>>(qout, (int)NBD);
  for (int q = 0; q < NQv; ++q)
    grvq_stage_kernel<<<TOKv / 128, 256, 0, stream>>>(resid, qout, cb, c2, lossA,
                                                      out + IDX_OFF, q);
  grvq_post_kernel<<<8192, 256, 0, stream>>>(qout, pow_, pob, out + Q_OFF);
  grvq_loss_kernel<<<1, 32, 0, stream>>>(lossA, out + LOSS_OFF);
}